// RoiAlign_27444841022136
// MI455X (gfx1250) — compile-verified
//
#include <hip/hip_runtime.h>
#include <hip/hip_bf16.h>

// ROI-align (multi-level FPN, TF crop_and_resize semantics) for MI455X/gfx1250.
//
// One workgroup (256 thr = 8 wave32) per box. Bilinear resize is expressed as
// a single matmul per output row on V_WMMA_F32_16X16X4_F32:
//     Out(ox, c) = sum_{k,r} A(ox, 2k+r) * F(y_r, col_k, c)
// i.e. BOTH x and y interpolation weights live in the sparse A matrix, so the
// B operand is raw feature data. Raw columns are staged global->LDS with
// GLOBAL_LOAD_ASYNC_TO_LDS_B128 (ASYNCcnt-tracked DMA, double-buffered so the
// next chunk streams in while WMMAs consume the current one). VALU only builds
// the A fragments. Memory-bound op: ~205MB stores + L2-resident gathers.

typedef float v2f __attribute__((ext_vector_type(2)));
typedef float v8f __attribute__((ext_vector_type(8)));

#define C_CH    256
#define RSTRIDE 272   // 256 + 16 floats pad: K-rows 2 apart land on disjoint bank groups
#define KCOLS   8     // source columns per chunk
#define NROWS   16    // K-rows per chunk: 2 per column (y0, y1)

__global__ __launch_bounds__(256)
void roi_align_wmma(const int* __restrict__ image_shape,
                    const float* __restrict__ boxes,
                    const float* __restrict__ p0, const float* __restrict__ p1,
                    const float* __restrict__ p2, const float* __restrict__ p3,
                    const float* __restrict__ p4,
                    float* __restrict__ out, int Nboxes)
{
    __shared__ float s_raw[2][NROWS * RSTRIDE];   // 2 * 16 * 272 * 4B = 34816 B
    __shared__ int   s_x0[16];
    __shared__ float s_wx[16];
    __shared__ float s_vx[16];
    __shared__ int   s_win[2];

    const int tid  = threadIdx.x;
    const int lane = tid & 31;
    const int wave = tid >> 5;
    const int bn   = blockIdx.x;            // flat b*N + n
    const int b    = bn / Nboxes;

    const float img_h = (float)image_shape[1];
    const float img_w = (float)image_shape[2];

    const float x1 = boxes[bn * 4 + 0];
    const float y1 = boxes[bn * 4 + 1];
    const float x2 = boxes[bn * 4 + 2];
    const float y2 = boxes[bn * 4 + 3];

    // ---- level selection: floor(1 + log2(sqrt(w*h)/224 + eps)), clip [0,4]
    const float bw = x2 - x1, bh = y2 - y1;
    const float size = sqrtf(bw * bh);
    float lv = floorf(1.0f + log2f(size / 224.0f + 1e-7f));
    lv = fminf(fmaxf(lv, 0.0f), 4.0f);
    const int level = (int)lv;

    const float* feat;
    int fdim;
    switch (level) {
        case 0:  feat = p0; fdim = 128; break;
        case 1:  feat = p1; fdim = 64;  break;
        case 2:  feat = p2; fdim = 32;  break;
        case 3:  feat = p3; fdim = 16;  break;
        default: feat = p4; fdim = 8;   break;
    }
    feat += (size_t)b * fdim * fdim * C_CH;
    const float fH = (float)fdim, fW = (float)fdim;

    // ---- normalized box, reference formulas
    const float y1n = y1 / img_h * fH / (fH - 1.0f);
    const float x1n = x1 / img_w * fW / (fW - 1.0f);
    const float y2n = (y2 / img_h * fH - 1.0f) / (fH - 1.0f);
    const float x2n = (x2 / img_w * fW - 1.0f) / (fW - 1.0f);

    // ---- x sampling grid (independent of oy): weights, validity, column window
    if (tid < 14) {
        const float tx = (float)tid * (1.0f / 13.0f);
        const float xs = (x1n + tx * (x2n - x1n)) * (fW - 1.0f);
        const int   vx = (xs >= 0.0f) && (xs <= fW - 1.0f);
        const float x0f = floorf(xs);
        const float wx  = xs - x0f;
        const int x0i = (int)fminf(fmaxf(x0f, 0.0f), fW - 1.0f);
        s_x0[tid] = x0i;
        s_wx[tid] = wx;
        s_vx[tid] = vx ? 1.0f : 0.0f;
    } else if (tid < 16) {          // pad rows m=14,15 of the A matrix -> zero weight
        s_x0[tid] = 0; s_wx[tid] = 0.0f; s_vx[tid] = 0.0f;
    }
    __syncthreads();
    if (tid == 0) {
        int lo = fdim - 1, hi = 0;
        for (int i = 0; i < 14; ++i) {
            const int a  = s_x0[i];
            const int a1 = min(a + 1, fdim - 1);
            lo = min(lo, a);
            hi = max(hi, a1);
        }
        s_win[0] = lo; s_win[1] = hi;
    }
    __syncthreads();
    const int xlo = s_win[0];
    const int xhi = s_win[1];

    // ---- per-lane A-fragment parameters (lane l covers matrix row m = l & 15)
    const int   rm  = lane & 15;
    const int   rx0 = s_x0[rm];
    const int   rx1 = min(rx0 + 1, fdim - 1);
    const float rwx = s_wx[rm];
    const float rvx = s_vx[rm];           // 0 for invalid xs and for pad rows 14,15
    const int   khalf = (lane >> 4) << 1; // lanes 0-15 -> K{0,1}; lanes 16-31 -> K{2,3}

    for (int oy = 0; oy < 14; ++oy) {
        const float ty  = (float)oy * (1.0f / 13.0f);
        const float ysv = (y1n + ty * (y2n - y1n)) * (fH - 1.0f);
        const bool  vy  = (ysv >= 0.0f) && (ysv <= fH - 1.0f);   // block-uniform
        const float y0f = floorf(ysv);
        const float fy  = ysv - y0f;
        const int y0i = (int)fminf(fmaxf(y0f, 0.0f), fH - 1.0f);
        const int y1i = min(y0i + 1, fdim - 1);

        v8f acc0 = {};   // channel tile  wave      (16 channels)
        v8f acc1 = {};   // channel tile  wave + 8

        if (vy) {
            // async DMA: stage K-rows 2k+r (column kc+k, source row y_r) into LDS.
            // per chunk: 16 K-rows x 256 ch x 4B = 16KB = 4 b128 issues per thread.
            auto prefetch = [&](int kc, int bufsel) {
                #pragma unroll
                for (int j = 0; j < 4; ++j) {
                    const int kidx = j * 4 + (tid >> 6);
                    const int col  = min(kc + (kidx >> 1), fdim - 1);
                    const int yr   = (kidx & 1) ? y1i : y0i;
                    const int c    = (tid & 63) * 4;
                    const float* src = feat + ((size_t)yr * fdim + col) * C_CH + c;
                    const unsigned la =
                        (unsigned)(uintptr_t)&s_raw[bufsel][kidx * RSTRIDE + c];
                    asm volatile("global_load_async_to_lds_b128 %0, %1, off"
                                 :: "v"(la),
                                    "v"((unsigned long long)(uintptr_t)src)
                                 : "memory");
                }
            };

            const int nchunk = ((xhi - xlo) >> 3) + 1;
            prefetch(xlo, 0);
            for (int i = 0; i < nchunk; ++i) {
                if (i + 1 < nchunk) {
                    // stream next chunk while this one is consumed; in-order
                    // async completion: <=4 outstanding => chunk i has landed.
                    prefetch(xlo + (i + 1) * KCOLS, (i + 1) & 1);
                    asm volatile("s_wait_asynccnt 0x4" ::: "memory");
                } else {
                    asm volatile("s_wait_asynccnt 0x0" ::: "memory");
                }
                __syncthreads();                  // all waves' DMA visible

                const float* raw = s_raw[i & 1];
                const int    kc  = xlo + i * KCOLS;

                #pragma unroll
                for (int ks = 0; ks < 4; ++ks) {
                    const int kidx0 = ks * 4 + khalf;        // even: y0 row
                    const int col   = kc + (kidx0 >> 1);
                    // sparse bilinear weight: x-weight in A.x/A.y split by y-row
                    const float xw = (((col == rx0) ? (1.0f - rwx) : 0.0f) +
                                      ((col == rx1) ? rwx          : 0.0f)) * rvx;
                    v2f A;
                    A.x = xw * (1.0f - fy);                  // K = kidx0   (y0)
                    A.y = xw * fy;                           // K = kidx0+1 (y1)

                    const int n = lane & 15;
                    v2f B0, B1;
                    B0.x = raw[kidx0       * RSTRIDE + wave * 16 + n];
                    B0.y = raw[(kidx0 + 1) * RSTRIDE + wave * 16 + n];
                    B1.x = raw[kidx0       * RSTRIDE + (wave + 8) * 16 + n];
                    B1.y = raw[(kidx0 + 1) * RSTRIDE + (wave + 8) * 16 + n];

                    acc0 = __builtin_amdgcn_wmma_f32_16x16x4_f32(
                        false, A, false, B0, (short)0, acc0, false, false);
                    acc1 = __builtin_amdgcn_wmma_f32_16x16x4_f32(
                        false, A, false, B1, (short)0, acc1, false, false);
                }
                __syncthreads();                  // buffer free for next prefetch
            }
        }

        // --- store: D VGPR r holds rows ox=r (lanes 0-15) and ox=r+8 (lanes 16-31)
        const float vyf = vy ? 1.0f : 0.0f;
        #pragma unroll
        for (int r = 0; r < 8; ++r) {
            const int ox = r + ((lane >> 4) << 3);
            if (ox < 14) {
                const size_t idx = (((size_t)bn * 14 + oy) * 14 + ox) * C_CH;
                out[idx + wave * 16 + (lane & 15)]       = vyf * acc0[r];
                out[idx + (wave + 8) * 16 + (lane & 15)] = vyf * acc1[r];
            }
        }
    }
}

extern "C" void kernel_launch(void* const* d_in, const int* in_sizes, int n_in,
                              void* d_out, int out_size, void* d_ws, size_t ws_size,
                              hipStream_t stream) {
    const int*   image_shape = (const int*)d_in[0];
    const float* boxes       = (const float*)d_in[1];
    // d_in[2] = scores (unused by the reference output)
    const float* p0 = (const float*)d_in[3];
    const float* p1 = (const float*)d_in[4];
    const float* p2 = (const float*)d_in[5];
    const float* p3 = (const float*)d_in[6];
    const float* p4 = (const float*)d_in[7];
    float* out = (float*)d_out;

    const int BN = in_sizes[2];                       // scores count = B*N
    const int Bv = in_sizes[3] / (128 * 128 * 256);   // p0 = [B,128,128,256]
    const int Nv = BN / Bv;

    roi_align_wmma<<<BN, 256, 0, stream>>>(image_shape, boxes,
                                           p0, p1, p2, p3, p4, out, Nv);
}